// SpectralGraphConv_19095424598064
// MI455X (gfx1250) — compile-verified
//
#include <hip/hip_runtime.h>
#include <hip/hip_bf16.h>

// SpectralGraphConv on gfx1250 (MI455X), f32 WMMA path.
//
// Only the first MODES=64 columns of the 8192x8192 eigenvector matrix matter:
//   xf = U[:, :64]^T @ X        (64 x 16, batches packed into columns)
//   of = mode-wise mix with w_real (w_imag never contributes: input is real)
//   out = U[:, :64] @ of        (8192 x 16)
// Both GEMMs run on V_WMMA_F32_16X16X4_F32 (f32 in, f32 out).

typedef __attribute__((ext_vector_type(2))) float v2f;
typedef __attribute__((ext_vector_type(8))) float v8f;

#define N_NODES 8192
#define CIN     8
#define COUT    8
#define MODES   64
#define KSPLIT  64   // split-K factor for the forward GFT

// ---------------------------------------------------------------------------
// Stage 1: xf[m, c'] = sum_n U[n, m] * x[b, n, c],  c' = b*8 + c  (64 x 16)
// grid = (4 M-tiles, KSPLIT), block = 32 (one wave).
// ---------------------------------------------------------------------------
__global__ void gft_forward_kernel(const float* __restrict__ U,
                                   const float* __restrict__ x,
                                   float* __restrict__ xf) {
    const int mtile = blockIdx.x;            // 0..3  -> modes m0..m0+15
    const int ks    = blockIdx.y;            // 0..KSPLIT-1
    const int lane  = threadIdx.x;           // 0..31
    const int m0    = mtile * 16;
    const int klen  = N_NODES / KSPLIT;      // 128 rows of n per wave
    const int nbeg  = ks * klen;

    const int half = lane >> 4;              // 0: K={0,1}, 1: K={2,3}
    const int r    = lane & 15;              // A row (mode) / B column (c')
    const int k0   = half * 2;
    const int b    = r >> 3;                 // batch for B column
    const int c    = r & 7;                  // channel for B column

    v8f acc = {};
#pragma unroll 4
    for (int n = nbeg; n < nbeg + klen; n += 4) {
        // A[m, k] = U[n+k, m0+m]   (A is the transposed eigenvector tile)
        v2f a;
        a.x = U[(size_t)(n + k0    ) * N_NODES + m0 + r];
        a.y = U[(size_t)(n + k0 + 1) * N_NODES + m0 + r];
        // B[k, c'] = x[b, n+k, c]
        v2f bb;
        bb.x = x[((size_t)b * N_NODES + (n + k0    )) * CIN + c];
        bb.y = x[((size_t)b * N_NODES + (n + k0 + 1)) * CIN + c];
        acc = __builtin_amdgcn_wmma_f32_16x16x4_f32(
            /*neg_a=*/false, a, /*neg_b=*/false, bb,
            /*c_mod=*/(short)0, acc, /*reuse_a=*/false, /*reuse_b=*/false);
    }

    // D layout: VGPR v, lane l -> M = v + (l/16)*8, N = l%16
#pragma unroll
    for (int v = 0; v < 8; ++v) {
        const int m = m0 + v + half * 8;
        atomicAdd(&xf[m * 16 + r], acc[v]);
    }
}

// ---------------------------------------------------------------------------
// Stage 2: of[m, j'] = sum_i xf[m, b*8+i] * w_real[i, j, m]   (64 x 16, tiny)
// ---------------------------------------------------------------------------
__global__ void spectral_mix_kernel(const float* __restrict__ xf,
                                    const float* __restrict__ wr,
                                    float* __restrict__ of) {
    const int t = blockIdx.x * blockDim.x + threadIdx.x;
    if (t >= MODES * 16) return;
    const int m  = t >> 4;
    const int jp = t & 15;
    const int b  = jp >> 3;
    const int j  = jp & 7;
    float s = 0.f;
#pragma unroll
    for (int i = 0; i < CIN; ++i)
        s += xf[m * 16 + (b * 8 + i)] * wr[(i * COUT + j) * MODES + m];
    of[m * 16 + jp] = s;
}

// ---------------------------------------------------------------------------
// Stage 3: out[b, n, j] = sum_{m<64} U[n, m] * of[m, j']   (8192 x 16)
// grid = 512 row-tiles, block = 32 (one wave), 16 WMMAs each.
// ---------------------------------------------------------------------------
__global__ void gft_inverse_kernel(const float* __restrict__ U,
                                   const float* __restrict__ of,
                                   float* __restrict__ out) {
    const int ntile = blockIdx.x;            // 0..511
    const int lane  = threadIdx.x;
    const int half  = lane >> 4;
    const int r     = lane & 15;
    const int k0    = half * 2;
    const int n     = ntile * 16 + r;        // A row = output node row

    v8f acc = {};
#pragma unroll
    for (int m = 0; m < MODES; m += 4) {
        // A[row, k] = U[n, m+k] -> contiguous float2 along the U row
        v2f a = *(const v2f*)&U[(size_t)n * N_NODES + m + k0];
        // B[k, j'] = of[m+k, j']
        v2f bb;
        bb.x = of[(m + k0    ) * 16 + r];
        bb.y = of[(m + k0 + 1) * 16 + r];
        acc = __builtin_amdgcn_wmma_f32_16x16x4_f32(
            false, a, false, bb, (short)0, acc, false, false);
    }

    const int b = r >> 3;
    const int j = r & 7;
#pragma unroll
    for (int v = 0; v < 8; ++v) {
        const int row = ntile * 16 + v + half * 8;
        out[((size_t)b * N_NODES + row) * COUT + j] = acc[v];
    }
}

extern "C" void kernel_launch(void* const* d_in, const int* in_sizes, int n_in,
                              void* d_out, int out_size, void* d_ws, size_t ws_size,
                              hipStream_t stream) {
    const float* x  = (const float*)d_in[0];   // (2, 8192, 8)
    const float* U  = (const float*)d_in[1];   // (8192, 8192)
    const float* wr = (const float*)d_in[2];   // (8, 8, 64)
    // d_in[3] = w_imag: provably unused (input spectrum is purely real)

    float* xf = (float*)d_ws;                  // 64*16 f32 split-K accumulator
    float* of = xf + MODES * 16;               // 64*16 f32 mixed spectrum
    float* out = (float*)d_out;                // (2, 8192, 8)

    hipMemsetAsync(xf, 0, MODES * 16 * sizeof(float), stream);

    gft_forward_kernel<<<dim3(4, KSPLIT), 32, 0, stream>>>(U, x, xf);
    spectral_mix_kernel<<<4, 256, 0, stream>>>(xf, wr, of);
    gft_inverse_kernel<<<512, 32, 0, stream>>>(U, of, out);
}